// YoloLayer_29858612642069
// MI455X (gfx1250) — compile-verified
//
#include <hip/hip_runtime.h>
#include <stdint.h>

#define G 152
#define GG (G * G)            // 23104, multiple of 4
#define NA 3
#define NCH 10                // 7 + NUM_CLASSES
#define POS_PER_BLK 256

__device__ __forceinline__ float fast_exp(float x) {
    return __builtin_amdgcn_exp2f(x * 1.44269504088896340736f);   // v_exp_f32
}
__device__ __forceinline__ float fast_sigmoid(float x) {
    return __builtin_amdgcn_rcpf(1.0f + fast_exp(-x));            // v_rcp_f32
}

__global__ __launch_bounds__(256)
void yolo_layer_kernel(const float* __restrict__ x,
                       const float* __restrict__ anchors,
                       const int*   __restrict__ img_size,
                       float* __restrict__ out) {
    __shared__ float lds_in[NCH * POS_PER_BLK];    // 10 KB, channel-major staging
    __shared__ float lds_out[POS_PER_BLK * NCH];   // 10 KB, output-major staging

    const int tid   = threadIdx.x;
    const int chunk = blockIdx.x;                  // spatial chunk (0..90)
    const int a     = blockIdx.y;                  // anchor
    const int b     = blockIdx.z;                  // batch
    const int pa    = b * NA + a;

    const int s_base = chunk * POS_PER_BLK;

    // ---------- Phase 1: async DMA global -> LDS (10 contiguous 1KB channel runs) ----------
    // 640 chunks of 16B; chunk i covers channel c = i/64, local floats (i%64)*4 .. +3
    for (int i = tid; i < (NCH * POS_PER_BLK) / 4; i += 256) {
        const int c  = i >> 6;                 // 64 x 16B chunks per channel
        const int s4 = (i & 63) << 2;          // local float offset within channel
        int s_glob = s_base + s4;
        if (s_glob >= GG) s_glob = 0;          // clamp tail chunks to plane start (unused data)
        const size_t plane = (size_t)(b * (NA * NCH) + a * NCH + c);
        const float* gptr  = x + plane * (size_t)GG + (size_t)s_glob;
        const uint64_t gaddr   = (uint64_t)(uintptr_t)gptr;
        const uint32_t ldsaddr = (uint32_t)(uintptr_t)(&lds_in[c * POS_PER_BLK + s4]);
        asm volatile("global_load_async_to_lds_b128 %0, %1, off th:TH_LOAD_NT"
                     :
                     : "v"(ldsaddr), "v"(gaddr)
                     : "memory");
    }
    asm volatile("s_wait_asynccnt 0" ::: "memory");
    __syncthreads();

    // ---------- Phase 2: per-position transform, write AoS rows into LDS ----------
    const float stride_f = (float)img_size[0] / (float)G;   // 4.0
    const float aw = anchors[a * 2 + 0];
    const float ah = anchors[a * 2 + 1];

    const int s = s_base + tid;
    if (s < GG) {
        const float gx = (float)(s % G);
        const float gy = (float)(s / G);

        const float v0 = lds_in[0 * POS_PER_BLK + tid];
        const float v1 = lds_in[1 * POS_PER_BLK + tid];
        const float v2 = lds_in[2 * POS_PER_BLK + tid];
        const float v3 = lds_in[3 * POS_PER_BLK + tid];
        const float v4 = lds_in[4 * POS_PER_BLK + tid];
        const float v5 = lds_in[5 * POS_PER_BLK + tid];
        const float v6 = lds_in[6 * POS_PER_BLK + tid];
        const float v7 = lds_in[7 * POS_PER_BLK + tid];
        const float v8 = lds_in[8 * POS_PER_BLK + tid];
        const float v9 = lds_in[9 * POS_PER_BLK + tid];

        float* row = &lds_out[tid * NCH];
        row[0] = (fast_sigmoid(v0) + gx) * stride_f;            // px * stride
        row[1] = (fast_sigmoid(v1) + gy) * stride_f;            // py * stride
        row[2] = fminf(fast_exp(v2), 1000.0f) * aw;             // pw*(aw/stride)*stride
        row[3] = fminf(fast_exp(v3), 1000.0f) * ah;             // ph*(ah/stride)*stride
        row[4] = v4;                                            // pim
        row[5] = v5;                                            // pre
        row[6] = fast_sigmoid(v6);                              // pconf
        row[7] = fast_sigmoid(v7);                              // pcls
        row[8] = fast_sigmoid(v8);
        row[9] = fast_sigmoid(v9);
    }
    __syncthreads();

    // ---------- Phase 3: async DMA LDS -> global, coalesced 16B/lane stores ----------
    int nvalid = GG - s_base;
    if (nvalid > POS_PER_BLK) nvalid = POS_PER_BLK;
    const int nvec = (nvalid * NCH) >> 2;                       // float4 count (640 or 160)
    const size_t out_base = ((size_t)pa * (size_t)GG + (size_t)s_base) * (size_t)NCH;

    for (int j = tid; j < nvec; j += 256) {
        const uint64_t gaddr   = (uint64_t)(uintptr_t)(out + out_base + (size_t)j * 4);
        const uint32_t ldsaddr = (uint32_t)(uintptr_t)(&lds_out[j * 4]);
        asm volatile("global_store_async_from_lds_b128 %0, %1, off th:TH_STORE_NT"
                     :
                     : "v"(gaddr), "v"(ldsaddr)
                     : "memory");
    }
    asm volatile("s_wait_asynccnt 0" ::: "memory");
}

extern "C" void kernel_launch(void* const* d_in, const int* in_sizes, int n_in,
                              void* d_out, int out_size, void* d_ws, size_t ws_size,
                              hipStream_t stream) {
    const float* x       = (const float*)d_in[0];
    const float* anchors = (const float*)d_in[1];
    const int*   imgsz   = (const int*)d_in[2];
    float* out = (float*)d_out;

    const int B = in_sizes[0] / (NA * NCH * GG);                // 32
    const int chunks = (GG + POS_PER_BLK - 1) / POS_PER_BLK;    // 91

    dim3 grid(chunks, NA, B);
    yolo_layer_kernel<<<grid, 256, 0, stream>>>(x, anchors, imgsz, out);
}